// SheafGluingCG_42906723287395
// MI455X (gfx1250) — compile-verified
//
#include <hip/hip_runtime.h>
#include <math.h>

// SheafGluingCG on MI455X (gfx1250, wave32).
// A = I + sum_e M_e^T M_e, M_e = [R_src(e) | -R_dst(e)] (8x16).
// Edge kernel: one edge per wave, 6x V_WMMA_F32_16X16X4_F32 per edge
// (4 chained for r = M q, 2 chained for M^T r), atomic scatter-add.
// HBM-bound (~205MB of R data per matvec), so WMMA shape/utilization is
// not the bottleneck; FP32 kept to match the reference exactly.

typedef __attribute__((ext_vector_type(2))) float v2f;
typedef __attribute__((ext_vector_type(8))) float v8f;

#define CG_EPS 1e-12f
#define CG_TOL 1e-6f

// ---------------------------------------------------------------------------
// Edge operator: out += sum over edges of M_e^T M_e gather(p). out must be
// pre-initialized to p (LAM == 1).
// One edge per wave. EXEC stays all-1 inside the wave-uniform live branch.
// ---------------------------------------------------------------------------
__global__ void edge_apply_wmma(const float* __restrict__ p,
                                const int* __restrict__ src,
                                const int* __restrict__ dst,
                                const float* __restrict__ Rs_g,
                                const float* __restrict__ Rd_g,
                                float* __restrict__ out,
                                int E, int M) {
  const int lane = threadIdx.x & 31;
  const int wave = threadIdx.x >> 5;
  const int eg = blockIdx.x * (blockDim.x >> 5) + wave;
  if (eg >= E) return;  // wave-uniform: whole wave exits together

  const int n = lane & 15;        // A-row / B-col / D-col index for this lane
  const bool hi = lane >= 16;     // upper lane half handles K+2, rows+8
  const int c = hi ? 2 : 0;       // K sub-offset per ISA f32 operand layout

  const int s = src[eg];
  const int t = dst[eg];
  const size_t eb = (size_t)eg * 64;
  const size_t mstride = (size_t)M * 8;

  // --- Step-1 A operand: Ms[a][j], a = n (rows 8..15 zero-padded) ---------
  // pairs for K-chunks k0 = 0,4,8,12; Ms[:,0:8]=R_src, Ms[:,8:16]=-R_dst
  float mA0, mA1, mA2, mA3, mA4, mA5, mA6, mA7;
  if (n < 8) {
    const float* Rs = Rs_g + eb + (size_t)n * 8;
    const float* Rd = Rd_g + eb + (size_t)n * 8;
    mA0 =  Rs[c];     mA1 =  Rs[c + 1];
    mA2 =  Rs[c + 4]; mA3 =  Rs[c + 5];
    mA4 = -Rd[c];     mA5 = -Rd[c + 1];
    mA6 = -Rd[c + 4]; mA7 = -Rd[c + 5];
  } else {
    mA0 = mA1 = mA2 = mA3 = mA4 = mA5 = mA6 = mA7 = 0.f;
  }

  // --- Step-1 B operand: q[j][b], b = n (cols 2..15 zero-padded) ----------
  float qB0, qB1, qB2, qB3, qB4, qB5, qB6, qB7;
  if (n < 2) {
    const float* ps = p + (size_t)n * mstride + (size_t)s * 8;
    const float* pt = p + (size_t)n * mstride + (size_t)t * 8;
    qB0 = ps[c];     qB1 = ps[c + 1];
    qB2 = ps[c + 4]; qB3 = ps[c + 5];
    qB4 = pt[c];     qB5 = pt[c + 1];
    qB6 = pt[c + 4]; qB7 = pt[c + 5];
  } else {
    qB0 = qB1 = qB2 = qB3 = qB4 = qB5 = qB6 = qB7 = 0.f;
  }

  // --- Step 1: r(8x2) = Ms(8x16) x q(16x2), K=16 via 4 chained WMMAs ------
  v8f acc = {0.f, 0.f, 0.f, 0.f, 0.f, 0.f, 0.f, 0.f};
  {
    v2f A; v2f Bv;
    A.x = mA0; A.y = mA1; Bv.x = qB0; Bv.y = qB1;   // k0 = 0
    acc = __builtin_amdgcn_wmma_f32_16x16x4_f32(false, A, false, Bv,
                                                (short)0, acc, false, false);
    A.x = mA2; A.y = mA3; Bv.x = qB2; Bv.y = qB3;   // k0 = 4
    acc = __builtin_amdgcn_wmma_f32_16x16x4_f32(false, A, false, Bv,
                                                (short)0, acc, false, false);
    A.x = mA4; A.y = mA5; Bv.x = qB4; Bv.y = qB5;   // k0 = 8
    acc = __builtin_amdgcn_wmma_f32_16x16x4_f32(false, A, false, Bv,
                                                (short)0, acc, false, false);
    A.x = mA6; A.y = mA7; Bv.x = qB6; Bv.y = qB7;   // k0 = 12
    acc = __builtin_amdgcn_wmma_f32_16x16x4_f32(false, A, false, Bv,
                                                (short)0, acc, false, false);
  }

  // Broadcast r[m][n] to every lane (row m lives in VGPR m, col in lane).
  float rsh0 = __shfl(acc[0], n, 32);
  float rsh1 = __shfl(acc[1], n, 32);
  float rsh2 = __shfl(acc[2], n, 32);
  float rsh3 = __shfl(acc[3], n, 32);
  float rsh4 = __shfl(acc[4], n, 32);
  float rsh5 = __shfl(acc[5], n, 32);
  float rsh6 = __shfl(acc[6], n, 32);
  float rsh7 = __shfl(acc[7], n, 32);

  // --- Step-2 A operand: (Ms^T)[j][k] = Ms[k][j], j = n, k pairs k0=0,4 ---
  float mT0, mT1, mT2, mT3;
  {
    const float* Rcol;
    float sg;
    if (n < 8) { Rcol = Rs_g + eb + n;        sg =  1.f; }
    else       { Rcol = Rd_g + eb + (n - 8);  sg = -1.f; }
    mT0 = sg * Rcol[(size_t)(c)     * 8];
    mT1 = sg * Rcol[(size_t)(c + 1) * 8];
    mT2 = sg * Rcol[(size_t)(c + 4) * 8];
    mT3 = sg * Rcol[(size_t)(c + 5) * 8];
  }

  // --- Step-2 B operand: B2[k][n] = r[k0 + c + k][n] ----------------------
  float b00 = hi ? rsh2 : rsh0;
  float b01 = hi ? rsh3 : rsh1;
  float b10 = hi ? rsh6 : rsh4;
  float b11 = hi ? rsh7 : rsh5;

  // --- Step 2: [c_src; c_dst](16x2) = Ms^T(16x8) x r(8x2), 2 WMMAs --------
  v8f acc2 = {0.f, 0.f, 0.f, 0.f, 0.f, 0.f, 0.f, 0.f};
  {
    v2f A; v2f Bv;
    A.x = mT0; A.y = mT1; Bv.x = b00; Bv.y = b01;   // k0 = 0
    acc2 = __builtin_amdgcn_wmma_f32_16x16x4_f32(false, A, false, Bv,
                                                 (short)0, acc2, false, false);
    A.x = mT2; A.y = mT3; Bv.x = b10; Bv.y = b11;   // k0 = 4
    acc2 = __builtin_amdgcn_wmma_f32_16x16x4_f32(false, A, false, Bv,
                                                 (short)0, acc2, false, false);
  }

  // --- Scatter: D2 rows 0..7 -> src node, rows 8..15 -> dst node ----------
  // lanes 0,1 hold rows 0..7 (cols b=0,1); lanes 16,17 hold rows 8..15.
  if (n < 2) {
    const int node = hi ? t : s;
    float* op = out + (size_t)n * mstride + (size_t)node * 8;
    atomicAdd(&op[0], acc2[0]);
    atomicAdd(&op[1], acc2[1]);
    atomicAdd(&op[2], acc2[2]);
    atomicAdd(&op[3], acc2[3]);
    atomicAdd(&op[4], acc2[4]);
    atomicAdd(&op[5], acc2[5]);
    atomicAdd(&op[6], acc2[6]);
    atomicAdd(&op[7], acc2[7]);
  }
}

// ---------------------------------------------------------------------------
// CG helper kernels
// ---------------------------------------------------------------------------
__global__ void copy_kernel(const float* __restrict__ in, float* __restrict__ out, int n) {
  int i = blockIdx.x * blockDim.x + threadIdx.x;
  if (i < n) out[i] = in[i];
}

__global__ void residual_kernel(const float* __restrict__ b, const float* __restrict__ Ab,
                                float* __restrict__ rr, float* __restrict__ p, int n) {
  int i = blockIdx.x * blockDim.x + threadIdx.x;
  if (i < n) {
    float v = b[i] - Ab[i];
    rr[i] = v;
    p[i] = v;
  }
}

__global__ void dot_kernel(const float* __restrict__ x, const float* __restrict__ y,
                           float* __restrict__ out, int n) {
  float s = 0.f;
  for (int i = blockIdx.x * blockDim.x + threadIdx.x; i < n;
       i += blockDim.x * gridDim.x)
    s += x[i] * y[i];
#pragma unroll
  for (int off = 16; off > 0; off >>= 1) s += __shfl_down(s, off, 32);
  if ((threadIdx.x & 31) == 0) atomicAdd(out, s);
}

// scal layout: [0,1]=rsold  [2,3]=pAp  [4,5]=rsnew  [6,7]=alpha
//              [8,9]=beta   [10]=done  [11]=upd (this-iteration active flag)
__global__ void init_scal_kernel(float* scal) {
  for (int i = 0; i < 16; ++i) scal[i] = 0.f;
}

__global__ void alpha_kernel(float* scal) {
  float done = scal[10];
  scal[6] = scal[0] / (scal[2] + CG_EPS);
  scal[7] = scal[1] / (scal[3] + CG_EPS);
  scal[11] = (done != 0.f) ? 0.f : 1.f;
  scal[4] = 0.f;  // clear rsnew accumulators for upcoming dot
  scal[5] = 0.f;
}

__global__ void beta_kernel(float* scal) {
  float done = scal[10];
  scal[8] = scal[4] / (scal[0] + CG_EPS);
  scal[9] = scal[5] / (scal[1] + CG_EPS);
  scal[11] = (done != 0.f) ? 0.f : 1.f;
  float mean = 0.5f * (scal[4] + scal[5]);
  if (done == 0.f) { scal[0] = scal[4]; scal[1] = scal[5]; }
  if (sqrtf(mean) < CG_TOL) scal[10] = 1.f;
  scal[2] = 0.f;  // clear pAp accumulators for next iteration
  scal[3] = 0.f;
}

__global__ void update_xr_kernel(float* __restrict__ x, float* __restrict__ rr,
                                 const float* __restrict__ p, const float* __restrict__ Ap,
                                 const float* __restrict__ scal, int md, int n) {
  int i = blockIdx.x * blockDim.x + threadIdx.x;
  if (i >= n) return;
  if (scal[11] == 0.f) return;
  float a = scal[6 + (i >= md ? 1 : 0)];
  x[i] += a * p[i];
  rr[i] -= a * Ap[i];
}

__global__ void update_p_kernel(float* __restrict__ p, const float* __restrict__ rr,
                                const float* __restrict__ scal, int md, int n) {
  int i = blockIdx.x * blockDim.x + threadIdx.x;
  if (i >= n) return;
  if (scal[11] == 0.f) return;
  float bt = scal[8 + (i >= md ? 1 : 0)];
  p[i] = rr[i] + bt * p[i];
}

// ---------------------------------------------------------------------------
extern "C" void kernel_launch(void* const* d_in, const int* in_sizes, int n_in,
                              void* d_out, int out_size, void* d_ws, size_t ws_size,
                              hipStream_t stream) {
  const float* c0 = (const float*)d_in[0];
  const int* src = (const int*)d_in[1];
  const int* dst = (const int*)d_in[2];
  const float* Rs = (const float*)d_in[3];
  const float* Rd = (const float*)d_in[4];

  const int N = in_sizes[0];   // B*M*d = 800000
  const int md = N / 2;        // M*d per batch
  const int M = md / 8;
  const int E = in_sizes[1];

  float* x = (float*)d_out;
  float* ws = (float*)d_ws;
  float* rr = ws;
  float* p = ws + (size_t)N;
  float* Ap = ws + 2 * (size_t)N;
  float* scal = ws + 3 * (size_t)N;

  const int TB = 256;
  const int gN = (N + TB - 1) / TB;
  const int gE = (E + 7) / 8;     // 8 waves (=8 edges) per block
  const int gDot = 512;

  init_scal_kernel<<<1, 1, 0, stream>>>(scal);

  // x0 = b = c0 ; r0 = b - A b ; p0 = r0 ; rsold = <r0,r0> per batch
  copy_kernel<<<gN, TB, 0, stream>>>(c0, x, N);
  copy_kernel<<<gN, TB, 0, stream>>>(c0, Ap, N);
  edge_apply_wmma<<<gE, TB, 0, stream>>>(c0, src, dst, Rs, Rd, Ap, E, M);
  residual_kernel<<<gN, TB, 0, stream>>>(c0, Ap, rr, p, N);
  dot_kernel<<<gDot, TB, 0, stream>>>(rr, rr, &scal[0], md);
  dot_kernel<<<gDot, TB, 0, stream>>>(rr + md, rr + md, &scal[1], md);

  for (int it = 0; it < 20; ++it) {
    // Ap = A p = p + sum_e M_e^T M_e p
    copy_kernel<<<gN, TB, 0, stream>>>(p, Ap, N);
    edge_apply_wmma<<<gE, TB, 0, stream>>>(p, src, dst, Rs, Rd, Ap, E, M);
    dot_kernel<<<gDot, TB, 0, stream>>>(p, Ap, &scal[2], md);
    dot_kernel<<<gDot, TB, 0, stream>>>(p + md, Ap + md, &scal[3], md);
    alpha_kernel<<<1, 1, 0, stream>>>(scal);
    update_xr_kernel<<<gN, TB, 0, stream>>>(x, rr, p, Ap, scal, md, N);
    dot_kernel<<<gDot, TB, 0, stream>>>(rr, rr, &scal[4], md);
    dot_kernel<<<gDot, TB, 0, stream>>>(rr + md, rr + md, &scal[5], md);
    beta_kernel<<<1, 1, 0, stream>>>(scal);
    update_p_kernel<<<gN, TB, 0, stream>>>(p, rr, scal, md, N);
  }

  (void)n_in; (void)out_size; (void)ws_size;
}